// DoubleQValueNet_37005438222966
// MI455X (gfx1250) — compile-verified
//
#include <hip/hip_runtime.h>
#include <math.h>

// ---------------------------------------------------------------------------
// DoubleQValueNet for MI455X (gfx1250, wave32).
// All GEMMs use V_WMMA_F32_16X16X4_F32 (exact fp32 matrix FMA on CDNA5).
// Each wave owns a 32x32 C tile: two A fragments share two B fragments per
// k-step -> 6 loads per 4 WMMAs. Segment sums use fp32 global atomics.
// ---------------------------------------------------------------------------

#define N_NODES 50000
#define N_EDGES 262144
#define E_SEP   524288
#define FDIM    128
#define NUM_SG  32768

typedef float v2f __attribute__((ext_vector_type(2)));
typedef float v8f __attribute__((ext_vector_type(8)));

__device__ __forceinline__ float lrelu(float x) { return x >= 0.0f ? x : 0.01f * x; }

// ---------------------------------------------------------------------------
// Generic dual-term WMMA GEMM:
//   C[M,N] = act( A1[idx1?][.]*rs1? @ W1  (+ A2[idx2?][.]*rs2? @ W2)
//                 (+ bias[col]) (+ t0[row]*tw0[col] + t1[row]*tw1[col]) )
// A row-major [*,K], W row-major [K,N], C row-major [M,N].
// M %16==0, N %32==0, K %4==0. 256 threads = 8 waves, wave = 32x32 tile
// (ragged last 16 rows handled by duplicating the lo fragment and
// discarding the hi stores -- keeps EXEC all-ones for WMMA).
// ---------------------------------------------------------------------------
__global__ __launch_bounds__(256) void k_gemm_wmma(
    const float* __restrict__ A1, const int* __restrict__ idx1, const float* __restrict__ rs1,
    const float* __restrict__ A2, const int* __restrict__ idx2, const float* __restrict__ rs2,
    const float* __restrict__ W1, const float* __restrict__ W2,
    const float* __restrict__ bias,
    const float* __restrict__ t0, const float* __restrict__ tw0,
    const float* __restrict__ t1, const float* __restrict__ tw1,
    float* __restrict__ C, int M, int N, int K, int dolrelu)
{
    const int lane = threadIdx.x & 31;
    const int wid  = threadIdx.x >> 5;
    const int l16  = lane & 15;
    const int half = lane >> 4;            // lanes 16-31: K+2/K+3 (A/B), M+8 (C)
    const int tilesN  = N >> 5;
    const int tilesM2 = (M + 31) >> 5;
    const int tile = blockIdx.x * 8 + wid;
    if (tile >= tilesM2 * tilesN) return;
    const int tm = tile / tilesN;
    const int tn = tile - tm * tilesN;
    const int row32 = tm << 5;
    const int col0  = (tn << 5) + l16;
    const bool hasHi = (row32 + 16) < M;

    v8f acc00 = {0.f,0.f,0.f,0.f,0.f,0.f,0.f,0.f};   // rows row32+0..15,  cols col0
    v8f acc01 = acc00;                                // rows lo,           cols col0+16
    v8f acc10 = acc00;                                // rows row32+16..31, cols col0
    v8f acc11 = acc00;                                // rows hi,           cols col0+16
    const int rLo = row32 + l16;
    const int rHiRaw = rLo + 16;
    const int rHi = hasHi ? rHiRaw : rLo;             // clamp: ragged tile reuses lo rows

    // ---- term 1 ----
    {
        const int   raL = idx1 ? idx1[rLo] : rLo;
        const int   raH = idx1 ? idx1[rHi] : rHi;
        const float scL = rs1 ? rs1[raL] : 1.0f;
        const float scH = rs1 ? rs1[raH] : 1.0f;
        const float* apL = A1 + (size_t)raL * K + 2 * half;
        const float* apH = A1 + (size_t)raH * K + 2 * half;
        const float* wp  = W1 + (size_t)(2 * half) * N + col0;
        #pragma unroll 4
        for (int k = 0; k < K; k += 4, apL += 4, apH += 4, wp += (size_t)4 * N) {
            v2f aL = *(const v2f*)apL; aL *= scL;     // A[r][k+2h], A[r][k+2h+1]
            v2f aH = *(const v2f*)apH; aH *= scH;
            v2f b0; b0.x = wp[0];  b0.y = wp[N];      // W[k+2h][c], W[k+2h+1][c]
            v2f b1; b1.x = wp[16]; b1.y = wp[N + 16];
            acc00 = __builtin_amdgcn_wmma_f32_16x16x4_f32(false, aL, false, b0, (short)0, acc00, false, false);
            acc01 = __builtin_amdgcn_wmma_f32_16x16x4_f32(false, aL, false, b1, (short)0, acc01, false, false);
            acc10 = __builtin_amdgcn_wmma_f32_16x16x4_f32(false, aH, false, b0, (short)0, acc10, false, false);
            acc11 = __builtin_amdgcn_wmma_f32_16x16x4_f32(false, aH, false, b1, (short)0, acc11, false, false);
        }
    }
    // ---- term 2 (optional) ----
    if (A2) {
        const int   raL = idx2 ? idx2[rLo] : rLo;
        const int   raH = idx2 ? idx2[rHi] : rHi;
        const float scL = rs2 ? rs2[raL] : 1.0f;
        const float scH = rs2 ? rs2[raH] : 1.0f;
        const float* apL = A2 + (size_t)raL * K + 2 * half;
        const float* apH = A2 + (size_t)raH * K + 2 * half;
        const float* wp  = W2 + (size_t)(2 * half) * N + col0;
        #pragma unroll 4
        for (int k = 0; k < K; k += 4, apL += 4, apH += 4, wp += (size_t)4 * N) {
            v2f aL = *(const v2f*)apL; aL *= scL;
            v2f aH = *(const v2f*)apH; aH *= scH;
            v2f b0; b0.x = wp[0];  b0.y = wp[N];
            v2f b1; b1.x = wp[16]; b1.y = wp[N + 16];
            acc00 = __builtin_amdgcn_wmma_f32_16x16x4_f32(false, aL, false, b0, (short)0, acc00, false, false);
            acc01 = __builtin_amdgcn_wmma_f32_16x16x4_f32(false, aL, false, b1, (short)0, acc01, false, false);
            acc10 = __builtin_amdgcn_wmma_f32_16x16x4_f32(false, aH, false, b0, (short)0, acc10, false, false);
            acc11 = __builtin_amdgcn_wmma_f32_16x16x4_f32(false, aH, false, b1, (short)0, acc11, false, false);
        }
    }

    // ---- epilogue: bias + rank-2 tail + leaky-relu + store ----
    float bc0 = 0.f, bc1 = 0.f;
    if (bias) { bc0 = bias[col0]; bc1 = bias[col0 + 16]; }
    float w0c0 = 0.f, w0c1 = 0.f, w1c0 = 0.f, w1c1 = 0.f;
    const bool hasTail = (t0 != nullptr);
    if (hasTail) { w0c0 = tw0[col0]; w0c1 = tw0[col0 + 16];
                   w1c0 = tw1[col0]; w1c1 = tw1[col0 + 16]; }
    #pragma unroll
    for (int j = 0; j < 8; ++j) {
        // C/D layout: VGPR j = M=j (lanes 0-15) / M=j+8 (lanes 16-31)
        const int rowL = row32 + j + 8 * half;
        float ta0 = 0.f, ta1 = 0.f;
        if (hasTail) {
            const float a0 = t0[rowL], a1 = t1[rowL];
            ta0 = a0 * w0c0 + a1 * w1c0;
            ta1 = a0 * w0c1 + a1 * w1c1;
        }
        float v0 = acc00[j] + bc0 + ta0;
        float v1 = acc01[j] + bc1 + ta1;
        if (dolrelu) { v0 = lrelu(v0); v1 = lrelu(v1); }
        float* crow = C + (size_t)rowL * N;
        crow[col0]      = v0;
        crow[col0 + 16] = v1;
    }
    if (hasHi) {
        #pragma unroll
        for (int j = 0; j < 8; ++j) {
            const int rowH = row32 + 16 + j + 8 * half;
            float ta0 = 0.f, ta1 = 0.f;
            if (hasTail) {
                const float a0 = t0[rowH], a1 = t1[rowH];
                ta0 = a0 * w0c0 + a1 * w1c0;
                ta1 = a0 * w0c1 + a1 * w1c1;
            }
            float v0 = acc10[j] + bc0 + ta0;
            float v1 = acc11[j] + bc1 + ta1;
            if (dolrelu) { v0 = lrelu(v0); v1 = lrelu(v1); }
            float* crow = C + (size_t)rowH * N;
            crow[col0]      = v0;
            crow[col0 + 16] = v1;
        }
    }
}

// ---- scatter: Out[dst[e]][:] += X[src[e]][:]   (128 cols, wave per edge) ----
__global__ __launch_bounds__(256) void k_scatter_add(
    const float* __restrict__ X, const int* __restrict__ sidx,
    const int* __restrict__ didx, float* __restrict__ Out, int nE)
{
    const int lane = threadIdx.x & 31;
    const int e = blockIdx.x * 8 + (threadIdx.x >> 5);
    if (e >= nE) return;
    const int s = sidx[e], d = didx[e];
    const float4 v = ((const float4*)(X + (size_t)s * FDIM))[lane];
    float* o = Out + (size_t)d * FDIM + lane * 4;
    atomicAdd(o + 0, v.x); atomicAdd(o + 1, v.y);
    atomicAdd(o + 2, v.z); atomicAdd(o + 3, v.w);
}

// ---- gather: Y[r][:] = X[idx[r]][:]  (128 cols, wave per row) ----
__global__ __launch_bounds__(256) void k_gather(
    const float* __restrict__ X, const int* __restrict__ idx,
    float* __restrict__ Y, int n)
{
    const int lane = threadIdx.x & 31;
    const int r = blockIdx.x * 8 + (threadIdx.x >> 5);
    if (r >= n) return;
    ((float4*)(Y + (size_t)r * FDIM))[lane] =
        ((const float4*)(X + (size_t)idx[r] * FDIM))[lane];
}

__global__ __launch_bounds__(256) void k_count(const int* __restrict__ idx, float* __restrict__ cnt, int n)
{
    const int i = blockIdx.x * 256 + threadIdx.x;
    if (i < n) atomicAdd(cnt + idx[i], 1.0f);
}

__global__ __launch_bounds__(256) void k_invmax1(float* __restrict__ c, int n)
{
    const int i = blockIdx.x * 256 + threadIdx.x;
    if (i < n) c[i] = 1.0f / fmaxf(c[i], 1.0f);
}

// ---- mean over groups of 8 rows: Y[g][c] = mean_r X[g*8+r][c] ----
__global__ __launch_bounds__(256) void k_pool8(const float* __restrict__ X, float* __restrict__ Y, int total)
{
    const int i = blockIdx.x * 256 + threadIdx.x;
    if (i >= total) return;
    const int g = i >> 7, c = i & 127;
    const float* xp = X + (size_t)g * 8 * FDIM + c;
    float s = 0.f;
    #pragma unroll
    for (int rr = 0; rr < 8; ++rr) s += xp[rr * FDIM];
    Y[i] = s * 0.125f;
}

// ---- side loss: acc += (sigmoid(ef[e].wp) - gt[e])^2   (wave per edge) ----
__global__ __launch_bounds__(256) void k_sideloss(
    const float* __restrict__ ef, const float* __restrict__ wp,
    const float* __restrict__ gt, float* __restrict__ acc, int nE)
{
    const int lane = threadIdx.x & 31;
    const int e = blockIdx.x * 8 + (threadIdx.x >> 5);
    if (e >= nE) return;
    const float4 a = ((const float4*)(ef + (size_t)e * FDIM))[lane];
    const float4 w = ((const float4*)wp)[lane];
    float p = a.x * w.x + a.y * w.y + a.z * w.z + a.w * w.w;
    #pragma unroll
    for (int off = 16; off > 0; off >>= 1) p += __shfl_xor(p, off, 32);
    if (lane == 0) {
        const float s = 1.0f / (1.0f + expf(-p));
        const float d = s - gt[e];
        atomicAdd(acc, d * d);
    }
}

// ---- value head output: q[r] = H[r][:512] . W3 + b3   (wave per row) ----
__global__ __launch_bounds__(256) void k_value_out(
    const float* __restrict__ H, const float* __restrict__ W3,
    const float* __restrict__ b3, float* __restrict__ q, int n)
{
    const int lane = threadIdx.x & 31;
    const int r = blockIdx.x * 8 + (threadIdx.x >> 5);
    if (r >= n) return;
    const float4* h4 = (const float4*)(H + (size_t)r * 512);
    const float4* w4 = (const float4*)W3;
    float s = 0.f;
    #pragma unroll
    for (int i = 0; i < 4; ++i) {
        const float4 a = h4[lane + 32 * i];
        const float4 w = w4[lane + 32 * i];
        s += a.x * w.x + a.y * w.y + a.z * w.z + a.w * w.w;
    }
    #pragma unroll
    for (int off = 16; off > 0; off >>= 1) s += __shfl_xor(s, off, 32);
    if (lane == 0) q[r] = s + b3[0];
}

__global__ void k_final_loss(const float* __restrict__ lacc, float* __restrict__ out)
{
    if (threadIdx.x == 0 && blockIdx.x == 0)
        out[0] = (lacc[0] + lacc[1]) * (0.25f / (float)N_EDGES);
}

// ---------------------------------------------------------------------------
// Input layout: top-level dict in insertion order; 'params' pytree-flattened
// (dict keys sorted alphabetically at every level).
// ---------------------------------------------------------------------------
enum {
    IN_NF = 0, IN_ACT, IN_ANG, IN_GT,
    P_G1_WA, P_G1_WS, P_G1_B,
    P_G2_WA, P_G2_WS, P_G2_B,
    P_Q1_WE, P_Q1_WM, P_Q1_WN, P_Q1_WP,
    P_Q2_WE, P_Q2_WM, P_Q2_WN, P_Q2_WP,
    P_V1_W1, P_V1_W2, P_V1_W3, P_V1_B1, P_V1_B2, P_V1_B3,
    P_V2_W1, P_V2_W2, P_V2_W3, P_V2_B1, P_V2_B2, P_V2_B3,
    IN_EI, IN_SUBG, IN_SEP, IN_POST
};

extern "C" void kernel_launch(void* const* d_in, const int* in_sizes, int n_in,
                              void* d_out, int out_size, void* d_ws, size_t ws_size,
                              hipStream_t stream)
{
    (void)in_sizes; (void)n_in; (void)out_size; (void)ws_size;

    const float* nf   = (const float*)d_in[IN_NF];
    const float* act  = (const float*)d_in[IN_ACT];
    const float* ang  = (const float*)d_in[IN_ANG];
    const float* gt   = (const float*)d_in[IN_GT];
    const int*   ei   = (const int*)d_in[IN_EI];          // [2, N_EDGES]
    const int*   subg = (const int*)d_in[IN_SUBG];        // [N_EDGES]
    const int*   sep  = (const int*)d_in[IN_SEP];         // [2, E_SEP]
    const int* ei_src = ei;
    const int* ei_dst = ei + N_EDGES;
    const int* sep0   = sep;
    const int* sep1   = sep + E_SEP;

    // ---- workspace carve-up (floats); heavy aliasing, total ~538 MB ----
    float* ws = (float*)d_ws;
    const size_t BIG = (size_t)N_EDGES * FDIM;            // 33,554,432 floats
    float* ef1    = ws;
    float* ef2    = ws + BIG;
    float* subA   = ws + 2 * BIG;
    float* aggE   = ws + 3 * BIG;
    // qgcnn temporaries alias subA (dead before gather into subA):
    float* hbuf   = subA;
    float* aggN   = subA + (size_t)N_NODES * FDIM;
    float* h2buf  = subA + (size_t)2 * N_NODES * FDIM;
    // value temporaries alias aggE (dead after gcnn iterations) and ef (subB):
    float* pooled = aggE;
    float* hv1    = aggE + (size_t)NUM_SG * FDIM;
    // small buffers
    float* invcntN = ws + 4 * BIG;
    float* invdegE = invcntN + N_NODES;
    float* lossacc = invdegE + N_EDGES;                   // 2 floats

    auto gemm = [&](const float* A1, const int* i1, const float* s1,
                    const float* A2, const int* i2, const float* s2,
                    const float* W1, const float* W2, const float* b,
                    const float* t0, const float* tw0,
                    const float* t1, const float* tw1,
                    float* Cp, int M, int N, int K, int actf) {
        const int tiles = ((M + 31) / 32) * (N / 32);
        k_gemm_wmma<<<(tiles + 7) / 8, 256, 0, stream>>>(
            A1, i1, s1, A2, i2, s2, W1, W2, b, t0, tw0, t1, tw1, Cp, M, N, K, actf);
    };

    // ---- degrees (shared by both nets) + loss accumulator ----
    hipMemsetAsync(invcntN, 0, (size_t)(N_NODES + N_EDGES + 2) * 4, stream);
    k_count<<<(N_EDGES + 255) / 256, 256, 0, stream>>>(ei_dst, invcntN, N_EDGES);
    k_count<<<(E_SEP + 255) / 256, 256, 0, stream>>>(sep1, invdegE, E_SEP);   // sg fwd dst
    k_count<<<(E_SEP + 255) / 256, 256, 0, stream>>>(sep0, invdegE, E_SEP);   // sg rev dst
    k_invmax1<<<(N_NODES + 255) / 256, 256, 0, stream>>>(invcntN, N_NODES);
    k_invmax1<<<(N_EDGES + 255) / 256, 256, 0, stream>>>(invdegE, N_EDGES);

    // ---- qgcnn x2: produce ef1 / ef2 + side-loss partials ----
    for (int net = 0; net < 2; ++net) {
        const float* We = (const float*)d_in[net ? P_Q2_WE : P_Q1_WE];  // [258,128]
        const float* Wm = (const float*)d_in[net ? P_Q2_WM : P_Q1_WM];  // [256,128]
        const float* Wn = (const float*)d_in[net ? P_Q2_WN : P_Q1_WN];  // [128,128]
        const float* wp = (const float*)d_in[net ? P_Q2_WP : P_Q1_WP];  // [128]
        float* ef = net ? ef2 : ef1;

        // h = lrelu(nf @ Wn)
        gemm(nf, nullptr, nullptr, nullptr, nullptr, nullptr,
             Wn, nullptr, nullptr, nullptr, nullptr, nullptr, nullptr,
             hbuf, N_NODES, FDIM, FDIM, 1);
        // agg = segment_mean(h[src] -> dst)
        hipMemsetAsync(aggN, 0, (size_t)N_NODES * FDIM * 4, stream);
        k_scatter_add<<<(N_EDGES + 7) / 8, 256, 0, stream>>>(hbuf, ei_src, ei_dst, aggN, N_EDGES);
        // h2 = lrelu(h @ Wm[0:128] + (agg*invcnt) @ Wm[128:256])
        gemm(hbuf, nullptr, nullptr, aggN, nullptr, invcntN,
             Wm, Wm + 128 * FDIM, nullptr, nullptr, nullptr, nullptr, nullptr,
             h2buf, N_NODES, FDIM, FDIM, 1);
        // ef = lrelu(h2[src]@We_top + h2[dst]@We_bot + act*We[256] + ang*We[257])
        gemm(h2buf, ei_src, nullptr, h2buf, ei_dst, nullptr,
             We, We + 128 * FDIM, nullptr,
             act, We + 256 * FDIM, ang, We + 257 * FDIM,
             ef, N_EDGES, FDIM, FDIM, 1);
        // side loss partial
        k_sideloss<<<(N_EDGES + 7) / 8, 256, 0, stream>>>(ef, wp, gt, lossacc + net, N_EDGES);
    }

    // ---- edge-graph GCN x2 (DEPTH=4) + value heads ----
    for (int net = 0; net < 2; ++net) {
        const float* Wa = (const float*)d_in[net ? P_G2_WA : P_G1_WA];
        const float* Gs = (const float*)d_in[net ? P_G2_WS : P_G1_WS];
        const float* gb = (const float*)d_in[net ? P_G2_B  : P_G1_B ];
        const float* V1 = (const float*)d_in[net ? P_V2_W1 : P_V1_W1];
        const float* V2 = (const float*)d_in[net ? P_V2_W2 : P_V1_W2];
        const float* V3 = (const float*)d_in[net ? P_V2_W3 : P_V1_W3];
        const float* vb1 = (const float*)d_in[net ? P_V2_B1 : P_V1_B1];
        const float* vb2 = (const float*)d_in[net ? P_V2_B2 : P_V1_B2];
        const float* vb3 = (const float*)d_in[net ? P_V2_B3 : P_V1_B3];
        float* ef   = net ? ef2 : ef1;
        float* subB = ef;                       // ef dead after gather -> reuse

        k_gather<<<(N_EDGES + 7) / 8, 256, 0, stream>>>(ef, subg, subA, N_EDGES);
        float* cur = subA, *nxt = subB;
        for (int it = 0; it < 4; ++it) {
            hipMemsetAsync(aggE, 0, (size_t)N_EDGES * FDIM * 4, stream);
            k_scatter_add<<<(E_SEP + 7) / 8, 256, 0, stream>>>(cur, sep0, sep1, aggE, E_SEP);
            k_scatter_add<<<(E_SEP + 7) / 8, 256, 0, stream>>>(cur, sep1, sep0, aggE, E_SEP);
            // nxt = lrelu(cur@Ws + (aggE*invdeg)@Wa + b)
            gemm(cur, nullptr, nullptr, aggE, nullptr, invdegE,
                 Gs, Wa, gb, nullptr, nullptr, nullptr, nullptr,
                 nxt, N_EDGES, FDIM, FDIM, 1);
            float* t = cur; cur = nxt; nxt = t;
        }
        // cur == subA after 4 iterations
        k_pool8<<<(NUM_SG * FDIM + 255) / 256, 256, 0, stream>>>(cur, pooled, NUM_SG * FDIM);
        // value MLP: 128 -> 512 -> 512 -> 1
        gemm(pooled, nullptr, nullptr, nullptr, nullptr, nullptr,
             V1, nullptr, vb1, nullptr, nullptr, nullptr, nullptr,
             hv1, NUM_SG, 512, FDIM, 1);
        float* hv2 = subB;                      // scratch iterate buffer is free now
        gemm(hv1, nullptr, nullptr, nullptr, nullptr, nullptr,
             V2, nullptr, vb2, nullptr, nullptr, nullptr, nullptr,
             hv2, NUM_SG, 512, 512, 1);
        k_value_out<<<(NUM_SG + 7) / 8, 256, 0, stream>>>(
            hv2, V3, vb3, (float*)d_out + (size_t)net * NUM_SG, NUM_SG);
    }

    k_final_loss<<<1, 32, 0, stream>>>(lossacc, (float*)d_out + 2 * NUM_SG);
}